// CCSDS123Compressor_23218593202870
// MI455X (gfx1250) — compile-verified
//
#include <hip/hip_runtime.h>
#include <stdint.h>

// ---------------------------------------------------------------------------
// CCSDS-123 adaptive sign-LMS predictor, CDNA5 (gfx1250) implementation.
//   image : [Z=202, Y*X=65536] f32     w0 : [202, 3] f32 (zeros)
//   out   : 6 sections of Z*65536 f32/i32:
//           pred | res | qres(=res) | mapped(i32) | samples(=image) | recon
// Kernel 1 (serial limit): 7 blocks x 1 wave32, lane = one band. TDM
//   (tensor_load_to_lds) double-buffers [35 band-rows x 128 t] tiles into LDS
//   (descriptor pad => conflict-free 132-float rows); recurrence in registers;
//   stores only pred+res to keep the issue stream lean.
// Kernel 2 (bandwidth limit): wide pointwise epilogue derives the other 4
//   sections with fully coalesced f32x4 traffic.
// ---------------------------------------------------------------------------

#define ZB      202
#define PN      65536        // pixels per band (raster order)
#define TC      128          // timesteps per TDM tile
#define NCHUNK  (PN / TC)    // 512
#define ROWS    35           // 32 output bands + 3 halo bands
#define RSTRIDE 132          // floats per LDS row: 128 data + 4 pad (TDM pads)
#define BUFSZ   (ROWS * RSTRIDE)
#define ZN      (ZB * PN)    // 13,238,272 elements per output section
#define MU_STEP 0.0009765625f
#define MAXV    32767.0f
#define MINV    (-32768.0f)

typedef __attribute__((ext_vector_type(4))) unsigned int u32x4;
typedef __attribute__((ext_vector_type(8))) int          i32x8;
typedef __attribute__((ext_vector_type(4))) int          i32x4;
typedef __attribute__((ext_vector_type(4))) float        f32x4;

// Issue one TDM tile load: tile [TC x tileRows] of f32, tensor row stride PN,
// destination LDS rows padded to RSTRIDE via the descriptor pad feature
// (pad_interval code 6 = every 128 dwords, pad_amount code 3 = 4 dwords).
__device__ __forceinline__ void tdm_load_tile(unsigned long long gaddr,
                                              unsigned int lds_byte_addr,
                                              int tileRows, int tensorRows) {
  u32x4 g0;
  g0.x = 1u;                                                // count=1, user desc
  g0.y = lds_byte_addr;                                     // lds_addr
  g0.z = (unsigned int)(gaddr & 0xFFFFFFFFull);             // global_addr[31:0]
  g0.w = (unsigned int)((gaddr >> 32) & 0x1FFFFFFull)       // global_addr[56:32]
       | (2u << 30);                                        // type = 2 (image)

  i32x8 g1;
  g1[0] = (2 << 16)            // data_size = 4 bytes
        | (1 << 20)            // pad_enable
        | (6 << 22)            // pad_interval: 128 dwords
        | (3 << 25);           // pad_amount:   4 dwords
  g1[1] = (int)((PN & 0xFFFF) << 16);                        // tensor_dim0 lo16
  g1[2] = (int)(((PN >> 16) & 0xFFFF)                        // tensor_dim0 hi16
        | ((tensorRows & 0xFFFF) << 16));                    // tensor_dim1 lo16
  g1[3] = (int)(((tensorRows >> 16) & 0xFFFF)                // tensor_dim1 hi16
        | ((TC & 0xFFFF) << 16));                            // tile_dim0 = 128
  g1[4] = (tileRows & 0xFFFF);                               // tile_dim1; dim2=0
  g1[5] = PN;                                                // dim0_stride lo32
  g1[6] = 0;                                                 // dim0_stride hi16
  g1[7] = 0;

  i32x4 z4 = {0, 0, 0, 0};
#if __has_builtin(__builtin_amdgcn_tensor_load_to_lds)
#if defined(__clang_major__) && (__clang_major__ >= 23)
  i32x8 z8 = {0, 0, 0, 0, 0, 0, 0, 0};
  __builtin_amdgcn_tensor_load_to_lds(g0, g1, z4, z4, z8, 0);
#else
  __builtin_amdgcn_tensor_load_to_lds(g0, g1, z4, z4, 0);
#endif
#else
  asm volatile("tensor_load_to_lds %0, %1, %2, %3"
               :: "s"(g0), "s"(g1), "s"(z4), "s"(z4) : "memory");
#endif
}

__global__ __launch_bounds__(32) void
ccsds123_scan_kernel(const float* __restrict__ img,
                     const float* __restrict__ w0in,
                     float* __restrict__ out) {
  __shared__ float lds[2][BUFSZ];

  const int wg    = blockIdx.x;          // 0..6, each owns 32 output bands
  const int lane  = threadIdx.x;         // 0..31
  const int zbase = wg * 32;
  const int z     = zbase + lane;        // this lane's band
  const bool act  = (z < ZB);

  const int zfirst     = (wg == 0) ? 0 : (zbase - 3);   // first band in tile
  const int rowShift   = (wg == 0) ? 3 : 0;             // skip halo rows in LDS
  int tileRows         = ROWS - rowShift;               // 32 or 35
  const int tensorRows = ZB - zfirst;                   // bands left in tensor
  if (tileRows > tensorRows) tileRows = tensorRows;     // never read past image

  // Bands -3..-1 are zero by spec: zero halo rows of both buffers for wg 0.
  if (wg == 0) {
    for (int i = lane; i < 3 * RSTRIDE; i += 32) {
      lds[0][i] = 0.0f;
      lds[1][i] = 0.0f;
    }
  }
  asm volatile("s_wait_dscnt 0" ::: "memory");

  const unsigned long long gbase =
      (unsigned long long)(uintptr_t)img +
      4ull * (unsigned long long)zfirst * (unsigned long long)PN;
  const unsigned int ldsb0 =
      (unsigned int)(uintptr_t)(&lds[0][0]) + (unsigned)(rowShift * RSTRIDE * 4);
  const unsigned int ldsb1 =
      (unsigned int)(uintptr_t)(&lds[1][0]) + (unsigned)(rowShift * RSTRIDE * 4);

  // Prime the double buffer.
  tdm_load_tile(gbase,             ldsb0, tileRows, tensorRows);
  tdm_load_tile(gbase + 4ull * TC, ldsb1, tileRows, tensorRows);

  float w0 = 0.0f, w1 = 0.0f, w2 = 0.0f;
  if (act) {
    w0 = w0in[z * 3 + 0];
    w1 = w0in[z * 3 + 1];
    w2 = w0in[z * 3 + 2];
  }

  // LDS rows: row r holds band (zbase - 3 + r). Lane needs bands z-3..z.
  const int r0 = (lane + 3) * RSTRIDE;   // band z   (sample)
  const int r1 = (lane + 2) * RSTRIDE;   // band z-1 (n[0])
  const int r2 = (lane + 1) * RSTRIDE;   // band z-2 (n[1])
  const int r3 = (lane + 0) * RSTRIDE;   // band z-3 (n[2])
  const int obase = z * PN;

  for (int c = 0; c < NCHUNK; ++c) {
    if (c + 1 < NCHUNK) __builtin_amdgcn_s_wait_tensorcnt((short)1);
    else                __builtin_amdgcn_s_wait_tensorcnt((short)0);

    const float* buf = &lds[c & 1][0];
    const int t0 = c * TC;

    for (int j = 0; j < TC; j += 4) {
      // All inputs for 4 steps: w-independent, fetched off the critical path.
      f32x4 sv = *(const f32x4*)(buf + r0 + j);
      f32x4 a1 = *(const f32x4*)(buf + r1 + j);
      f32x4 a2 = *(const f32x4*)(buf + r2 + j);
      f32x4 a3 = *(const f32x4*)(buf + r3 + j);

      f32x4 P, R;
#pragma unroll
      for (int k = 0; k < 4; ++k) {
        const float s  = sv[k];
        const float n1 = a1[k], n2 = a2[k], n3 = a3[k];
        // Critical path: 3 fma -> clip -> sub -> sign-select -> 3 fma (w).
        const float d    = fmaf(w2, n3, fmaf(w1, n2, w0 * n1));
        const float pred = fminf(fmaxf(d, MINV), MAXV);
        const float res  = s - pred;
        const float u    = (res > 0.0f) ? MU_STEP
                         : ((res < 0.0f) ? -MU_STEP : 0.0f);
        w0 = fmaf(u, n1, w0);
        w1 = fmaf(u, n2, w1);
        w2 = fmaf(u, n3, w2);
        P[k] = pred;
        R[k] = res;
      }

      if (act) {
        const int o = obase + t0 + j;
        *(f32x4*)(out + 0 * ZN + o) = P;    // predictions
        *(f32x4*)(out + 1 * ZN + o) = R;    // residuals
      }
    }

    // All LDS reads of this buffer must retire before the TDM overwrites it.
    asm volatile("s_wait_dscnt 0" ::: "memory");
    if (c + 2 < NCHUNK) {
      tdm_load_tile(gbase + 4ull * (unsigned long long)(c + 2) * TC,
                    (c & 1) ? ldsb1 : ldsb0, tileRows, tensorRows);
    }
  }
}

// Pointwise epilogue: derive the remaining 4 sections at HBM bandwidth.
//   qres = res; mapped = bijection(round_ne(res)); samples = image;
//   recon = pred + res.
__global__ __launch_bounds__(256) void
ccsds123_epilogue_kernel(const float* __restrict__ img,
                         float* __restrict__ out) {
  const int i = (blockIdx.x * 256 + threadIdx.x) * 4;
  if (i >= ZN) return;

  const f32x4 P = *(const f32x4*)(out + 0 * ZN + i);
  const f32x4 R = *(const f32x4*)(out + 1 * ZN + i);
  const f32x4 S = *(const f32x4*)(img + i);

  f32x4 C;
  i32x4 M;
#pragma unroll
  for (int k = 0; k < 4; ++k) {
    C[k] = P[k] + R[k];
    const int q = __float2int_rn(R[k]);             // round-half-even
    M[k] = (q >= 0) ? (2 * q) : (-2 * q - 1);
  }

  *(f32x4*)(out + 2 * ZN + i)        = R;   // quantized_residuals
  *(i32x4*)((int*)out + 3 * ZN + i)  = M;   // mapped_indices (int32)
  *(f32x4*)(out + 4 * ZN + i)        = S;   // sample_representatives
  *(f32x4*)(out + 5 * ZN + i)        = C;   // reconstructed
}

extern "C" void kernel_launch(void* const* d_in, const int* in_sizes, int n_in,
                              void* d_out, int out_size, void* d_ws, size_t ws_size,
                              hipStream_t stream) {
  (void)in_sizes; (void)n_in; (void)out_size; (void)d_ws; (void)ws_size;
  const float* img = (const float*)d_in[0];
  const float* w0  = (const float*)d_in[1];
  float* out       = (float*)d_out;

  // 7 blocks x 32 lanes = 224 lanes >= 202 bands (full band-level parallelism).
  hipLaunchKernelGGL(ccsds123_scan_kernel, dim3(7), dim3(32), 0, stream,
                     img, w0, out);
  // ZN / (256*4) = 12928 exactly.
  hipLaunchKernelGGL(ccsds123_epilogue_kernel, dim3(ZN / 1024), dim3(256), 0,
                     stream, img, out);
}